// L1AttnF_12670153523838
// MI455X (gfx1250) — compile-verified
//
#include <hip/hip_runtime.h>
#include <stdint.h>

typedef __attribute__((ext_vector_type(4))) unsigned int v4u;
typedef __attribute__((ext_vector_type(8))) int          v8i;
typedef __attribute__((ext_vector_type(4))) int          v4i;

#define N_CTX    512
#define N_HEADS  8
#define WIDTH    64
#define TILE     64
#define ROWSTRIDE 68   /* 64 dwords + 4 dwords TDM-inserted padding per row */
#define EPS_F    0.001f

// Truncate a generic pointer to a __shared__ object to its LDS byte offset.
// (Flat LDS-aperture addresses carry the wave-relative LDS offset in the low
//  32 bits per CDNA5 ISA aperture mapping.)
__device__ __forceinline__ unsigned lds_offset(const void* p) {
  return (unsigned)(unsigned long long)p;
}

// Issue one TDM 2D tile load: 64 rows x 64 f32, row stride = N_HEADS*WIDTH
// elements in global memory, padded to 68-dword rows in LDS.
__device__ __forceinline__ void tdm_load_tile(const float* gaddr, unsigned ldsoff) {
  const uint64_t ga = (uint64_t)(uintptr_t)gaddr;

  // ---- D# group 0 (128b): count | lds_addr | global_addr | type=2 ----
  v4u g0;
  g0.x = 1u;                                        // count=1 (valid), no gather
  g0.y = ldsoff;                                    // LDS byte address
  g0.z = (unsigned)(ga & 0xFFFFFFFFu);              // global_addr[31:0]
  g0.w = (unsigned)((ga >> 32) & 0x01FFFFFFu)       // global_addr[56:32]
       | (2u << 30);                                // type = 2 ("image")

  // ---- D# group 1 (256b) ----
  const unsigned td0   = WIDTH;                     // tensor_dim0 (elements)
  const unsigned td1   = N_CTX;                     // tensor_dim1 (rows)
  const unsigned tile0 = WIDTH;                     // tile_dim0
  const unsigned tile1 = TILE;                      // tile_dim1 (rows per tile)
  const unsigned str0  = N_HEADS * WIDTH;           // tensor_dim0_stride (elements)

  v8i g1;
  g1[0] = (int)((2u << 16)      /* data_size = 4 bytes      */
              | (1u << 20)      /* pad_enable               */
              | (5u << 22)      /* pad_interval: 64 dwords  */
              | (3u << 25));    /* pad_amount:   4 dwords   */
  g1[1] = (int)((td0 & 0xFFFFu) << 16);             // tensor_dim0[15:0] @ bit48
  g1[2] = (int)((td0 >> 16) | ((td1 & 0xFFFFu) << 16));
  g1[3] = (int)((td1 >> 16) | (tile0 << 16));
  g1[4] = (int)(tile1);                             // tile_dim2 = 0
  g1[5] = (int)str0;                                // tensor_dim0_stride[31:0]
  g1[6] = 0;                                        // stride0 hi | stride1 lo
  g1[7] = 0;                                        // stride1 hi

  // 2D tensor: groups 2/3 unused -> zeros; trailing 8-dword group zeroed too.
  v4i gz4 = (v4i)(0);
  v8i gz8 = (v8i)(0);

  __builtin_amdgcn_tensor_load_to_lds(g0, g1, gz4, gz4, gz8, 0);
}

__global__ __launch_bounds__(256)
void l1attn_tdm_kernel(const float* __restrict__ q,
                       const float* __restrict__ k,
                       float* __restrict__ out) {
  __shared__ float sq[TILE * ROWSTRIDE];   // [t][w], padded rows
  __shared__ float sk[TILE * ROWSTRIDE];   // [s][w], padded rows

  const int t0 = blockIdx.x * TILE;
  const int s0 = blockIdx.y * TILE;
  const int h  = blockIdx.z;

  // One wave drives the Tensor Data Mover for both tiles.
  if (threadIdx.x < 32) {
    tdm_load_tile(q + (size_t)t0 * (N_HEADS * WIDTH) + (size_t)h * WIDTH, lds_offset(sq));
    tdm_load_tile(k + (size_t)s0 * (N_HEADS * WIDTH) + (size_t)h * WIDTH, lds_offset(sk));
    __builtin_amdgcn_s_wait_tensorcnt(0);
  }
  __syncthreads();

  // 256 threads -> 16x16 grid of 4x4 register tiles covering 64x64 outputs.
  const int tid = threadIdx.x;
  const int ti  = (tid & 15) << 2;   // t offset within tile
  const int si  = (tid >> 4) << 2;   // s offset within tile

  float acc[4][4];
#pragma unroll
  for (int a = 0; a < 4; ++a)
#pragma unroll
    for (int b = 0; b < 4; ++b) acc[a][b] = 0.0f;

  for (int w = 0; w < WIDTH; w += 4) {
    float4 qv[4], kv[4];
#pragma unroll
    for (int i = 0; i < 4; ++i)
      qv[i] = *(const float4*)&sq[(ti + i) * ROWSTRIDE + w];
#pragma unroll
    for (int i = 0; i < 4; ++i)
      kv[i] = *(const float4*)&sk[(si + i) * ROWSTRIDE + w];

#pragma unroll
    for (int a = 0; a < 4; ++a) {
#pragma unroll
      for (int b = 0; b < 4; ++b) {
        acc[a][b] += fabsf(qv[b].x - kv[a].x)
                   + fabsf(qv[b].y - kv[a].y)
                   + fabsf(qv[b].z - kv[a].z)
                   + fabsf(qv[b].w - kv[a].w);
      }
    }
  }

  const float scale = 0.125f;  // 1/sqrt(64)
#pragma unroll
  for (int a = 0; a < 4; ++a) {
    float4 r;
    r.x = 1.0f / (EPS_F + acc[a][0] * scale);
    r.y = 1.0f / (EPS_F + acc[a][1] * scale);
    r.z = 1.0f / (EPS_F + acc[a][2] * scale);
    r.w = 1.0f / (EPS_F + acc[a][3] * scale);
    const size_t row = ((size_t)h * N_CTX + (size_t)(s0 + si + a)) * N_CTX;
    *(float4*)&out[row + (size_t)(t0 + ti)] = r;
  }
}

extern "C" void kernel_launch(void* const* d_in, const int* in_sizes, int n_in,
                              void* d_out, int out_size, void* d_ws, size_t ws_size,
                              hipStream_t stream) {
  const float* q = (const float*)d_in[0];
  const float* k = (const float*)d_in[1];
  float* out = (float*)d_out;

  dim3 grid(N_CTX / TILE, N_CTX / TILE, N_HEADS);  // 8 x 8 x 8 = 512 blocks
  l1attn_tdm_kernel<<<grid, 256, 0, stream>>>(q, k, out);
}